// BktModel_9242769621138
// MI455X (gfx1250) — compile-verified
//
#include <hip/hip_runtime.h>
#include <stdint.h>

// BKT forward-backward, chunked associative-scan formulation for MI455X.
// Phase 1: per-(row,chunk) 2x2 logsumexp-semiring products P_c.
// Phase 2: per-row scan over chunk products -> chunk-entry alpha, chunk-exit beta.
// Phase 3: per-(row,chunk) replay: forward (preds, posts) + backward (smoothed).
// Data staging: wave-cooperative global_load_async_to_lds_b128 (gfx1250 async
// engine), 512B-contiguous per issue, padded LDS layout (<=2-way bank conflict).
// Transcendentals: hardware v_exp_f32/v_log_f32 via __expf/__logf (1-3 ulp);
// normalized outputs are differences of correlated lse terms so error cancels.

#define B_DIM 8192
#define T_DIM 1024
#define CH 16
#define CNUM (T_DIM / CH)          // 64 chunks per row
#define CPW 32                     // chunks staged per wave (one row per wave-pair)
#define OBS_STRIDE_DW 36           // 32 dw data + 4 pad (144B, 16B aligned)
#define CORR_STRIDE_DW 20          // 16 dw data + 4 pad (80B, 16B aligned)
#define WAVE_LDS_DW (CPW * (OBS_STRIDE_DW + CORR_STRIDE_DW))   // 1792 dw = 7168B
#define BLOCK 256
#define WAVES_PER_BLOCK (BLOCK / 32)

// lse2(a,b) = log(e^a + e^b); softplus(x) = lse2(x, 0).
// Hardware-transcendental versions: ~9 VALU ops each (2 trans).
__device__ __forceinline__ float lse2(float a, float b) {
    return fmaxf(a, b) + __logf(1.0f + __expf(-fabsf(a - b)));
}
__device__ __forceinline__ float softplusf(float x) {
    return fmaxf(x, 0.0f) + __logf(1.0f + __expf(-fabsf(x)));
}

// Stage this wave's 32 chunks of obs (4KB) and corr (2KB) into padded LDS via
// the gfx1250 async global->LDS engine. Region is wave-private, so only
// s_wait_asynccnt is required (no workgroup barrier).
__device__ __forceinline__ void stage_chunks(const float* __restrict__ obs,
                                             const int* __restrict__ corr,
                                             int b, int c0, uint32_t* smem) {
    const int lane = threadIdx.x & 31;
    const int wid  = threadIdx.x >> 5;
    uint32_t* wbase = smem + wid * WAVE_LDS_DW;
    const uint32_t obsLds  = (uint32_t)(uintptr_t)wbase;       // generic->LDS offset
    const uint32_t corrLds = obsLds + CPW * OBS_STRIDE_DW * 4;
    const float* gobs  = obs  + (size_t)b * (T_DIM * 2) + (size_t)c0 * (CH * 2);
    const int*   gcorr = corr + (size_t)b * T_DIM       + (size_t)c0 * CH;
#pragma unroll
    for (int it = 0; it < 8; ++it) {            // 256 x 16B transfers, 8 per lane
        uint32_t q  = (uint32_t)(it * 32 + lane);
        uint64_t ga = (uint64_t)(uintptr_t)gobs + (uint64_t)q * 16u;
        uint32_t la = obsLds + (q >> 3) * (OBS_STRIDE_DW * 4) + (q & 7u) * 16u;
        asm volatile("global_load_async_to_lds_b128 %0, %1, off"
                     :: "v"(la), "v"(ga) : "memory");
    }
#pragma unroll
    for (int it = 0; it < 4; ++it) {            // 128 x 16B transfers, 4 per lane
        uint32_t q  = (uint32_t)(it * 32 + lane);
        uint64_t ga = (uint64_t)(uintptr_t)gcorr + (uint64_t)q * 16u;
        uint32_t la = corrLds + (q >> 2) * (CORR_STRIDE_DW * 4) + (q & 3u) * 16u;
        asm volatile("global_load_async_to_lds_b128 %0, %1, off"
                     :: "v"(la), "v"(ga) : "memory");
    }
    asm volatile("s_wait_asynccnt 0x0" ::: "memory");
}

// -------- Phase 1: chunk products P_c = A_{t15} (x) ... (x) A_{t0} --------
__global__ __launch_bounds__(BLOCK) void bkt_chunk_products(
        const int* __restrict__ corr, const float* __restrict__ dyn,
        const float* __restrict__ obs, float* __restrict__ Pws) {
    __shared__ __align__(16) uint32_t smem[WAVES_PER_BLOCK * WAVE_LDS_DW];
    const int tid = blockIdx.x * BLOCK + threadIdx.x;
    const int b = tid >> 6;
    const int c = tid & 63;
    const int c0 = c & ~(CPW - 1);
    stage_chunks(obs, corr, b, c0, smem);

    const int wid = threadIdx.x >> 5;
    const int cw  = c & (CPW - 1);
    const uint32_t* mobs  = smem + wid * WAVE_LDS_DW + cw * OBS_STRIDE_DW;
    const uint32_t* mcorr = smem + wid * WAVE_LDS_DW + CPW * OBS_STRIDE_DW + cw * CORR_STRIDE_DW;

    const float d0 = dyn[b * 3 + 0], d1 = dyn[b * 3 + 1];
    const float sd0 = softplusf(d0), sd1 = softplusf(d1);
    const float lt00 = -sd0, lt10 = d0 - sd0, lt01 = d1 - sd1, lt11 = -sd1;

    float P00 = 0.f, P01 = 0.f, P10 = 0.f, P11 = 0.f;
#pragma unroll
    for (int t = 0; t < CH; ++t) {
        float o0 = __uint_as_float(mobs[2 * t]);
        float o1 = __uint_as_float(mobs[2 * t + 1]);
        int   y  = (int)mcorr[t];
        float sp0 = softplusf(o0), sp1 = softplusf(o1);
        float py0 = y ? (o0 - sp0) : (-sp0);       // log P(y|h=0)
        float py1 = y ? (-sp1)     : (o1 - sp1);   // log P(y|h=1)
        float A00 = lt00 + py0, A01 = lt01 + py1;
        float A10 = lt10 + py0, A11 = lt11 + py1;
        if (t == 0) { P00 = A00; P01 = A01; P10 = A10; P11 = A11; }
        else {
            float n00 = lse2(A00 + P00, A01 + P10);
            float n01 = lse2(A00 + P01, A01 + P11);
            float n10 = lse2(A10 + P00, A11 + P10);
            float n11 = lse2(A10 + P01, A11 + P11);
            P00 = n00; P01 = n01; P10 = n10; P11 = n11;
        }
    }
    ((float4*)Pws)[tid] = make_float4(P00, P01, P10, P11);
}

// -------- Phase 2: per-row scans over chunk products --------
__global__ __launch_bounds__(256) void bkt_scan(
        const float* __restrict__ dyn, const float* __restrict__ Pws,
        float* __restrict__ Aws, float* __restrict__ Gws) {
    const int b = blockIdx.x * 256 + threadIdx.x;
    const float d2 = dyn[b * 3 + 2];
    const float sd2 = softplusf(d2);
    float a0 = -sd2, a1 = d2 - sd2;                // log_alpha0
    const float4* P4 = (const float4*)Pws;
    float2* A2 = (float2*)Aws;
    float2* G2 = (float2*)Gws;
    for (int cc = 0; cc < CNUM; ++cc) {            // alpha at chunk entry
        int idx = b * CNUM + cc;
        A2[idx] = make_float2(a0, a1);
        float4 P = P4[idx];
        float n0 = lse2(P.x + a0, P.y + a1);       // P (x) a
        float n1 = lse2(P.z + a0, P.w + a1);
        a0 = n0; a1 = n1;
    }
    float g0 = 0.f, g1 = 0.f;                      // beta at chunk exit (t = cE-1)
    for (int cc = CNUM - 1; cc >= 0; --cc) {
        int idx = b * CNUM + cc;
        G2[idx] = make_float2(g0, g1);
        float4 P = P4[idx];
        float n0 = lse2(P.x + g0, P.z + g1);       // P^T (x) g
        float n1 = lse2(P.y + g0, P.w + g1);
        g0 = n0; g1 = n1;
    }
}

// -------- Phase 3: within-chunk replay, emit all three outputs --------
__global__ __launch_bounds__(BLOCK) void bkt_within_chunk(
        const int* __restrict__ corr, const float* __restrict__ dyn,
        const float* __restrict__ obs, const float* __restrict__ Aws,
        const float* __restrict__ Gws, float* __restrict__ out) {
    __shared__ __align__(16) uint32_t smem[WAVES_PER_BLOCK * WAVE_LDS_DW];
    const int tid = blockIdx.x * BLOCK + threadIdx.x;
    const int b = tid >> 6;
    const int c = tid & 63;
    const int c0 = c & ~(CPW - 1);
    stage_chunks(obs, corr, b, c0, smem);

    const int wid = threadIdx.x >> 5;
    const int cw  = c & (CPW - 1);
    const uint32_t* mobs  = smem + wid * WAVE_LDS_DW + cw * OBS_STRIDE_DW;
    const uint32_t* mcorr = smem + wid * WAVE_LDS_DW + CPW * OBS_STRIDE_DW + cw * CORR_STRIDE_DW;

    const float d0 = dyn[b * 3 + 0], d1 = dyn[b * 3 + 1];
    const float sd0 = softplusf(d0), sd1 = softplusf(d1);
    const float lt00 = -sd0, lt10 = d0 - sd0, lt01 = d1 - sd1, lt11 = -sd1;

    float2 av = ((const float2*)Aws)[tid];
    float a0 = av.x, a1 = av.y;

    float2* preds  = (float2*)out + (size_t)b * T_DIM + (size_t)c * CH;
    float2* posts  = (float2*)out + (size_t)B_DIM * T_DIM     + (size_t)b * T_DIM + (size_t)c * CH;
    float2* smooth = (float2*)out + (size_t)2 * B_DIM * T_DIM + (size_t)b * T_DIM + (size_t)c * CH;

    float PP0[CH], PP1[CH], QQ0[CH], QQ1[CH];
#pragma unroll
    for (int t = 0; t < CH; ++t) {
        float o0 = __uint_as_float(mobs[2 * t]);
        float o1 = __uint_as_float(mobs[2 * t + 1]);
        int   y  = (int)mcorr[t];
        float sp0 = softplusf(o0), sp1 = softplusf(o1);
        float lo00 = -sp0, lo01 = o0 - sp0, lo10 = o1 - sp1, lo11 = -sp1;
        // predictive p(y_t | y_<t), normalized
        float u0 = lse2(lo00 + a0, lo10 + a1);
        float u1 = lse2(lo01 + a0, lo11 + a1);
        float n  = lse2(u0, u1);
        preds[t] = make_float2(u0 - n, u1 - n);
        // emission of observed symbol, filtered joint P(h_t, y_1:t)
        float py0 = y ? lo01 : lo00;
        float py1 = y ? lo11 : lo10;
        float p0 = py0 + a0, p1 = py1 + a1;
        posts[t] = make_float2(p0, p1);
        PP0[t] = p0; PP1[t] = p1; QQ0[t] = py0; QQ1[t] = py1;
        // alpha update
        a0 = lse2(lt00 + p0, lt01 + p1);
        a1 = lse2(lt10 + p0, lt11 + p1);
    }

    float2 gv = ((const float2*)Gws)[tid];
    float b0 = gv.x, b1 = gv.y;                    // beta at t = chunk end - 1
#pragma unroll
    for (int t = CH - 1; t >= 0; --t) {
        float s0 = PP0[t] + b0, s1 = PP1[t] + b1;
        float n = lse2(s0, s1);
        smooth[t] = make_float2(s0 - n, s1 - n);
        float nb0 = lse2(QQ0[t] + b0 + lt00, QQ1[t] + b1 + lt10);
        float nb1 = lse2(QQ0[t] + b0 + lt01, QQ1[t] + b1 + lt11);
        b0 = nb0; b1 = nb1;
    }
}

extern "C" void kernel_launch(void* const* d_in, const int* in_sizes, int n_in,
                              void* d_out, int out_size, void* d_ws, size_t ws_size,
                              hipStream_t stream) {
    (void)in_sizes; (void)n_in; (void)out_size; (void)ws_size;
    const int*   corr = (const int*)d_in[0];
    const float* dyn  = (const float*)d_in[1];
    const float* obs  = (const float*)d_in[2];
    float* out = (float*)d_out;
    // workspace carve: P (8MB) | alphaStart (4MB) | betaExit (4MB)
    float* Pws = (float*)d_ws;
    float* Aws = Pws + (size_t)B_DIM * CNUM * 4;
    float* Gws = Aws + (size_t)B_DIM * CNUM * 2;

    const int nThreads = B_DIM * CNUM;   // 524288
    bkt_chunk_products<<<nThreads / BLOCK, BLOCK, 0, stream>>>(corr, dyn, obs, Pws);
    bkt_scan<<<B_DIM / 256, 256, 0, stream>>>(dyn, Pws, Aws, Gws);
    bkt_within_chunk<<<nThreads / BLOCK, BLOCK, 0, stream>>>(corr, dyn, obs, Aws, Gws, out);
}